// GNNEncoder_38019050504275
// MI455X (gfx1250) — compile-verified
//
#include <hip/hip_runtime.h>
#include <hip/hip_bf16.h>

// ---------------------------------------------------------------------------
// R-GCN encoder for MI455X (gfx1250, wave32).
//   h = x @ emb_w + emb_b
//   4 layers: out = h@root_w + root_b; out += scatter_max((h@conv_w)[src], dst)
//             (x4 relations); h = relu(out)
//   d_out = segment_sum(h, batch) -> [64,128]
// GEMMs use V_WMMA_F32_16X16X4_F32 (full fp32). Scatter-max uses
// order-preserving u32 encoding + global_atomic_max_u32 (deterministic).
// ---------------------------------------------------------------------------

#define N_NODES  100000
#define N_EDGES  500000
#define N_REL    4
#define N_LAYERS 4
#define IN_FEAT  64
#define EMB      128
#define N_GRAPHS 64
#define NB       ((size_t)N_NODES * EMB)      // elems per node-feature buffer

typedef float v2f __attribute__((ext_vector_type(2)));
typedef float v8f __attribute__((ext_vector_type(8)));

// order-preserving fp32 <-> u32 map (monotone: a<b  <=>  enc(a)<enc(b))
__device__ __forceinline__ unsigned enc_f32(float x) {
    unsigned u = __float_as_uint(x);
    return (u & 0x80000000u) ? ~u : (u | 0x80000000u);
}
__device__ __forceinline__ float dec_f32(unsigned u) {
    unsigned v = (u & 0x80000000u) ? (u & 0x7FFFFFFFu) : ~u;
    return __uint_as_float(v);
}
#define ENC_NEG_INF 0x007FFFFFu   // enc(-inf): "no incoming message" sentinel

// ---------------------------------------------------------------------------
// WMMA fp32 GEMM:  C[nrows,128] = A[nrows,K] * W[K,128] (+ bias)
// Block = 128 threads = 4 waves; block tile = 64 rows x 128 cols.
// Wave tile = 16 rows x 128 cols -> 8 v8f accumulators, K/4 wmma steps.
// LDS staging with padded strides (130, K+2) to avoid 64-bank conflicts.
// ---------------------------------------------------------------------------
template <int K>
__global__ __launch_bounds__(128)
void gemm_wmma_f32(const float* __restrict__ A, const float* __restrict__ W,
                   const float* __restrict__ bias, float* __restrict__ C,
                   int nrows) {
    constexpr int WS = 130;      // padded W row stride (words): 130 % 64 = 2
    constexpr int AS = K + 2;    // padded A row stride (even -> b64 aligned)
    __shared__ float Wlds[K * WS];
    __shared__ float Alds[64 * AS];

    const int tid  = threadIdx.x;
    const int lane = tid & 31;
    const int wave = tid >> 5;
    const int rb   = blockIdx.x * 64;

    // stage W[K][128] -> LDS (coalesced, padded rows)
    for (int i = tid; i < K * 128; i += 128) {
        int k = i >> 7, c = i & 127;
        Wlds[k * WS + c] = W[i];
    }
    // stage A tile (64 rows x K), zero-fill out-of-range rows
    for (int i = tid; i < 64 * K; i += 128) {
        int row = i / K, kk = i - row * K;
        int gr  = rb + row;
        float v = 0.0f;
        if (gr < nrows) v = A[(size_t)gr * K + kk];
        Alds[row * AS + kk] = v;
    }
    __syncthreads();

    const int half = lane >> 4;            // 0: lanes 0-15, 1: lanes 16-31
    const int l15  = lane & 15;
    const int arow = wave * 16 + l15;      // this lane's A row within the tile

    float bfrag[8];
#pragma unroll
    for (int n0 = 0; n0 < 8; ++n0)
        bfrag[n0] = bias ? bias[n0 * 16 + l15] : 0.0f;

    v8f acc[8];
    v8f zero = {};
#pragma unroll
    for (int n0 = 0; n0 < 8; ++n0) acc[n0] = zero;

    for (int k0 = 0; k0 < K; k0 += 4) {
        // A frag (16x4 f32): lane<16 holds (M=l15, K=k0..k0+1),
        //                    lane>=16 holds (M=l15, K=k0+2..k0+3)
        const int kk = k0 + half * 2;
        v2f a = *(const v2f*)&Alds[arow * AS + kk];
#pragma unroll
        for (int n0 = 0; n0 < 8; ++n0) {
            // B frag (4x16 f32): VGPR0 = row kk, VGPR1 = row kk+1, N = l15
            v2f b;
            b.x = Wlds[kk * WS + n0 * 16 + l15];
            b.y = Wlds[(kk + 1) * WS + n0 * 16 + l15];
            acc[n0] = __builtin_amdgcn_wmma_f32_16x16x4_f32(
                false, a, false, b, (short)0, acc[n0], false, false);
        }
    }

    // C/D layout: VGPR i -> row (wave*16 + half*8 + i), col n0*16 + l15
    const int rbase = rb + wave * 16 + half * 8;
#pragma unroll
    for (int i = 0; i < 8; ++i) {
        int row = rbase + i;
        if (row < nrows) {
#pragma unroll
            for (int n0 = 0; n0 < 8; ++n0)
                C[(size_t)row * 128 + n0 * 16 + l15] = acc[n0][i] + bfrag[n0];
        }
    }
}

// ---------------------------------------------------------------------------
// fill agg with encoded -inf sentinel (uint4 stores)
// ---------------------------------------------------------------------------
__global__ __launch_bounds__(256)
void fill_u32(unsigned* __restrict__ p, unsigned v, int n4) {
    int i = blockIdx.x * 256 + threadIdx.x;
    if (i < n4) ((uint4*)p)[i] = make_uint4(v, v, v, v);
}

// ---------------------------------------------------------------------------
// scatter-max: one wave per edge; lane handles 4 features (float4 gather,
// 4 non-returning global_atomic_max_u32). Edge indices are wave-uniform.
// ---------------------------------------------------------------------------
__global__ __launch_bounds__(256)
void scatter_max(const float* __restrict__ t, const int* __restrict__ ei,
                 int nedges, unsigned* __restrict__ agg) {
    int idx = blockIdx.x * 256 + threadIdx.x;
    int e   = idx >> 5;
    int f   = (idx & 31) * 4;
    if (e >= nedges) return;
    int src = ei[e];               // row 0 of edge_index
    int dst = ei[nedges + e];      // row 1 of edge_index
    float4 v = *(const float4*)(t + (size_t)src * 128 + f);
    unsigned* d = agg + (size_t)dst * 128 + f;
    atomicMax(d + 0, enc_f32(v.x));
    atomicMax(d + 1, enc_f32(v.y));
    atomicMax(d + 2, enc_f32(v.z));
    atomicMax(d + 3, enc_f32(v.w));
}

// ---------------------------------------------------------------------------
// out += decode(agg) (sentinel -> 0); final relation also applies ReLU and
// writes h (next layer's input), fusing the extra pass.
// ---------------------------------------------------------------------------
__global__ __launch_bounds__(256)
void combine(float* __restrict__ out, const unsigned* __restrict__ agg,
             float* __restrict__ h, int n4, int do_relu) {
    int i = blockIdx.x * 256 + threadIdx.x;
    if (i >= n4) return;
    uint4  a = ((const uint4*)agg)[i];
    float4 o = ((const float4*)out)[i];
    float4 s;
    s.x = o.x + ((a.x == ENC_NEG_INF) ? 0.0f : dec_f32(a.x));
    s.y = o.y + ((a.y == ENC_NEG_INF) ? 0.0f : dec_f32(a.y));
    s.z = o.z + ((a.z == ENC_NEG_INF) ? 0.0f : dec_f32(a.z));
    s.w = o.w + ((a.w == ENC_NEG_INF) ? 0.0f : dec_f32(a.w));
    if (do_relu) {
        s.x = fmaxf(s.x, 0.0f); s.y = fmaxf(s.y, 0.0f);
        s.z = fmaxf(s.z, 0.0f); s.w = fmaxf(s.w, 0.0f);
        ((float4*)h)[i] = s;
    } else {
        ((float4*)out)[i] = s;
    }
}

// ---------------------------------------------------------------------------
// global_add_pool: batch is sorted, so each graph is a contiguous node
// segment. One block per graph, one thread per feature, fixed-order serial
// accumulation -> bitwise deterministic (no fp32 atomicAdd).
// ---------------------------------------------------------------------------
__device__ __forceinline__ int lower_bound_i32(const int* a, int n, int v) {
    int lo = 0, hi = n;
    while (lo < hi) { int m = (lo + hi) >> 1; if (a[m] < v) lo = m + 1; else hi = m; }
    return lo;
}

__global__ __launch_bounds__(128)
void pool_sum(const float* __restrict__ h, const int* __restrict__ batch,
              float* __restrict__ out, int nnodes) {
    int g = blockIdx.x;
    int f = threadIdx.x;
    int s = lower_bound_i32(batch, nnodes, g);
    int e = lower_bound_i32(batch, nnodes, g + 1);
    float acc = 0.0f;
    for (int n = s; n < e; ++n) acc += h[(size_t)n * 128 + f];
    out[(size_t)g * 128 + f] = acc;
}

// ---------------------------------------------------------------------------
extern "C" void kernel_launch(void* const* d_in, const int* in_sizes, int n_in,
                              void* d_out, int out_size, void* d_ws, size_t ws_size,
                              hipStream_t stream) {
    (void)in_sizes; (void)n_in; (void)out_size; (void)ws_size;
    const float* x      = (const float*)d_in[0];
    const int*   ei[4]  = {(const int*)d_in[1], (const int*)d_in[2],
                           (const int*)d_in[3], (const int*)d_in[4]};
    const int*   batch  = (const int*)d_in[5];
    const float* emb_w  = (const float*)d_in[6];
    const float* emb_b  = (const float*)d_in[7];
    const float* root_w = (const float*)d_in[8];
    const float* root_b = (const float*)d_in[9];
    const float* conv_w = (const float*)d_in[10];

    // workspace: h | out | t | agg  (each 100000*128 f32 = 51.2 MB)
    float*    h   = (float*)d_ws;
    float*    out = h + NB;
    float*    t   = out + NB;
    unsigned* agg = (unsigned*)(t + NB);

    const int gemm_blocks = (N_NODES + 63) / 64;       // 1563
    const int n4          = (int)(NB / 4);             // 3.2M float4
    const int fill_blocks = (n4 + 255) / 256;          // 12500
    const int scat_blocks = (N_EDGES * 32 + 255) / 256; // 62500

    // embedding: h = x @ emb_w + emb_b      (K = 64)
    gemm_wmma_f32<IN_FEAT><<<gemm_blocks, 128, 0, stream>>>(
        x, emb_w, emb_b, h, N_NODES);

    for (int l = 0; l < N_LAYERS; ++l) {
        // out = h @ root_w[l] + root_b[l]
        gemm_wmma_f32<EMB><<<gemm_blocks, 128, 0, stream>>>(
            h, root_w + (size_t)l * EMB * EMB, root_b + (size_t)l * EMB,
            out, N_NODES);

        for (int r = 0; r < N_REL; ++r) {
            // t = h @ conv_w[l][r]
            gemm_wmma_f32<EMB><<<gemm_blocks, 128, 0, stream>>>(
                h, conv_w + ((size_t)(l * N_REL + r)) * EMB * EMB, nullptr,
                t, N_NODES);
            // agg = enc(-inf)
            fill_u32<<<fill_blocks, 256, 0, stream>>>(agg, ENC_NEG_INF, n4);
            // agg[dst] = max(agg[dst], t[src])
            scatter_max<<<scat_blocks, 256, 0, stream>>>(t, ei[r], N_EDGES, agg);
            // out += dec(agg); last relation: h = relu(out + dec(agg))
            combine<<<fill_blocks, 256, 0, stream>>>(out, agg, h, n4,
                                                     (r == N_REL - 1) ? 1 : 0);
        }
    }

    // d_out[g] = sum of h over nodes with batch==g
    pool_sum<<<N_GRAPHS, 128, 0, stream>>>(h, batch, (float*)d_out, N_NODES);
}